// BlockCore_10651518894817
// MI455X (gfx1250) — compile-verified
//
#include <hip/hip_runtime.h>

typedef float v2f __attribute__((ext_vector_type(2)));
typedef float v8f __attribute__((ext_vector_type(8)));
typedef unsigned int u32x4 __attribute__((ext_vector_type(4)));
typedef int i32x8 __attribute__((ext_vector_type(8)));
typedef int i32x4 __attribute__((ext_vector_type(4)));

#define ROWS 16384          // B*T = 4*4096
#define NCOL 4099           // row stride / last-dim size
#define BLOCK_REGION 4096   // 1024 blocks * 4
#define LDS_STRIDE 132      // 128 + 4 pad -> conflict-free column reads
#define NCHUNK 32           // 4096 / 128

// One workgroup (256 threads = 8 wave32) owns 16 rows; loops over 32 chunks of
// 128 columns. Input staging: TDM tensor_load_to_lds (2D tile 128x16, HW pad
// interval 128 dwords / amount 4 dwords -> LDS stride 132), double-buffered so
// the DMA of chunk k+1 overlaps compute of chunk k. Compute: per-wave 16x16
// tile via 4x V_WMMA_F32_16X16X4_F32 (K=4 == block size). Output: D -> lds_y
// -> coalesced non-temporal global stores.
__global__ __launch_bounds__(256) void block_diag_wmma_kernel(
    const float* __restrict__ x,
    const float* __restrict__ blocks,
    float* __restrict__ y)
{
    __shared__ float lds_x[2][16 * LDS_STRIDE];
    __shared__ float lds_y[16 * LDS_STRIDE];

    const int tid  = threadIdx.x;
    const int lane = tid & 31;      // wave32
    const int wave = tid >> 5;      // 0..7
    const int row0 = blockIdx.x * 16;

    const int n    = lane & 15;     // A: row M ; B/D: column N
    const int half = lane >> 4;     // selects K pair {0,1} vs {2,3}

    // ---- TDM D# group1 (constant): data_size=4B(code 2), pad_enable,
    // pad_interval=128 dwords (code 6), pad_amount=4 dwords (code 3),
    // tensor_dim0=4099, tensor_dim1=16384, tile_dim0=128, tile_dim1=16,
    // tensor_dim0_stride=4099 ----
    i32x8 g1;
    g1[0] = (2 << 16) | (1 << 20) | (6 << 22) | (3 << 25);
    g1[1] = (int)((4099u & 0xFFFFu) << 16);                    // dim0 lo16 -> [31:16]
    g1[2] = (int)((4099u >> 16) | ((16384u & 0xFFFFu) << 16)); // dim0 hi | dim1 lo
    g1[3] = (int)((16384u >> 16) | (128u << 16));              // dim1 hi | tile_dim0
    g1[4] = 16;                                                // tile_dim1 (tile_dim2=0)
    g1[5] = 4099;                                              // dim0_stride lo32
    g1[6] = 0;                                                 // dim0_stride hi16
    g1[7] = 0;
    const i32x4 gz4 = {0, 0, 0, 0};                            // groups 2/3 unused (2D)
    const i32x8 gz8 = {0, 0, 0, 0, 0, 0, 0, 0};

    const unsigned lds_base0 = (unsigned)(size_t)(void*)&lds_x[0][0];
    const unsigned lds_base1 = (unsigned)(size_t)(void*)&lds_x[1][0];
    const unsigned long long gbase =
        (unsigned long long)(size_t)x + (unsigned long long)row0 * NCOL * 4ull;

    // prologue: DMA chunk 0 into buffer 0
    if (wave == 0) {
        u32x4 g0;
        g0[0] = 1u;                                   // count=1 (valid descriptor)
        g0[1] = lds_base0;
        g0[2] = (unsigned)gbase;
        g0[3] = (unsigned)(gbase >> 32) | (2u << 30); // type=2 ("image")
        __builtin_amdgcn_tensor_load_to_lds(g0, g1, gz4, gz4, gz8, 0);
    }

    for (int chunk = 0; chunk < NCHUNK; ++chunk) {
        const int cur  = chunk & 1;
        const int col0 = chunk * 128;

        if (wave == 0) {
            if (chunk + 1 < NCHUNK) {
                // prefetch next chunk into the other buffer, then wait for the
                // previous DMA (in-order per wave => cnt<=1 means it's done)
                u32x4 g0;
                unsigned long long ga = gbase + (unsigned long long)(col0 + 128) * 4ull;
                g0[0] = 1u;
                g0[1] = cur ? lds_base0 : lds_base1;
                g0[2] = (unsigned)ga;
                g0[3] = (unsigned)(ga >> 32) | (2u << 30);
                __builtin_amdgcn_tensor_load_to_lds(g0, g1, gz4, gz4, gz8, 0);
                __builtin_amdgcn_s_wait_tensorcnt(1);
            } else {
                __builtin_amdgcn_s_wait_tensorcnt(0);
            }
        }
        __syncthreads();

        const float* __restrict__ bx = &lds_x[cur][0];

        // ---- per-wave 16x16 output tile (4 blocks) ----
        const int ctile = wave * 16;            // local column base
        const int gblk  = (col0 + ctile) >> 2;  // first of 4 global block indices

        // For this lane's output column n, only block (n>>2) contributes.
        // blocks[b][j][k] at b*16 + j*4 + k ; j = n&3 ; k = 2*half + v
        const int   myblk = gblk + (n >> 2);
        const float bv0 = blocks[myblk * 16 + (n & 3) * 4 + 2 * half + 0];
        const float bv1 = blocks[myblk * 16 + (n & 3) * 4 + 2 * half + 1];

        v8f acc = {};
        #pragma unroll
        for (int i = 0; i < 4; ++i) {
            // A fragment: 16x4 slice of x at local cols [ctile+4i, ctile+4i+4)
            // layout: lane(M=n), VGPR v holds K = 2*half + v
            v2f a;
            a.x = bx[n * LDS_STRIDE + ctile + 4 * i + 2 * half + 0];
            a.y = bx[n * LDS_STRIDE + ctile + 4 * i + 2 * half + 1];
            // B fragment: blocks[gblk+i]^T in columns 4i..4i+3, zero elsewhere
            v2f b;
            b.x = ((n >> 2) == i) ? bv0 : 0.0f;
            b.y = ((n >> 2) == i) ? bv1 : 0.0f;
            acc = __builtin_amdgcn_wmma_f32_16x16x4_f32(
                /*neg_a=*/false, a, /*neg_b=*/false, b,
                /*c_mod=*/(short)0, acc, /*reuse_a=*/false, /*reuse_b=*/false);
        }

        // D layout: VGPR v -> row (v + 8*half), column n
        #pragma unroll
        for (int v = 0; v < 8; ++v) {
            lds_y[(v + 8 * half) * LDS_STRIDE + ctile + n] = acc[v];
        }
        __syncthreads();

        // ---- cooperative coalesced store (b32, NT) ----
        #pragma unroll
        for (int it = 0; it < 8; ++it) {
            int idx = tid + it * 256;
            int r = idx >> 7;
            int c = idx & 127;
            __builtin_nontemporal_store(lds_y[r * LDS_STRIDE + c],
                                        &y[(size_t)(row0 + r) * NCOL + col0 + c]);
        }
        __syncthreads();
    }
}

// y[row, 4096+j] = x[row, 4096+j] * diag[j], j in 0..2
__global__ __launch_bounds__(256) void remainder_kernel(
    const float* __restrict__ x,
    const float* __restrict__ diag,
    float* __restrict__ y)
{
    int t = blockIdx.x * blockDim.x + threadIdx.x;
    int row = t / 3;
    int j = t - row * 3;
    if (row < ROWS) {
        size_t off = (size_t)row * NCOL + BLOCK_REGION + j;
        y[off] = x[off] * diag[j];
    }
}

extern "C" void kernel_launch(void* const* d_in, const int* in_sizes, int n_in,
                              void* d_out, int out_size, void* d_ws, size_t ws_size,
                              hipStream_t stream) {
    const float* x      = (const float*)d_in[0];  // (4,4096,4099) f32
    const float* blocks = (const float*)d_in[1];  // (1024,4,4)    f32
    const float* diag   = (const float*)d_in[2];  // (3,)          f32
    float* y = (float*)d_out;                     // (4,4096,4099) f32

    block_diag_wmma_kernel<<<dim3(ROWS / 16), dim3(256), 0, stream>>>(x, blocks, y);

    int rem_elems = ROWS * 3;
    remainder_kernel<<<dim3((rem_elems + 255) / 256), dim3(256), 0, stream>>>(x, diag, y);
}